// Apply3DDispField_5506148074186
// MI455X (gfx1250) — compile-verified
//
#include <hip/hip_runtime.h>

// 3D displacement-field warp, trilinear with round-based anchor (faithful to
// the TF/JAX reference, including the out-of-[0,1] "extrapolation" weights
// the round() anchor produces past the .5 boundary).
//
// img (2,160,160,160,1) fp32, disp (2,160,160,160,3) fp32 -> out fp32.
// Memory-bound: ~164MB unique traffic -> ~7us floor at 23.3 TB/s HBM.
// WMMA/TDM inapplicable (per-lane data-dependent gather, ~1.5 flop/B); we
// lean on gfx1250's 192MB L2 (whole working set resident across replays)
// and global_prefetch_b8 to hide gather latency behind coordinate math.
//
// Launch: blockDim.x = 160 (one z-row, exactly 5 wave32), grid (160,160,2):
// no div/mod index decomposition, i/j grid coords are wave-uniform.

#define H_DIM 160
#define MAXC  160            // padded index range [0,160]; 160 == zero pad
#define SCALE 159.0f         // 0.5*(x+1)*(max-1), max-1 = 159

// Bit-exact jnp.linspace(-1,1,160):
//   step = fl(i/159);  out = (-1)*(1-step) + 1*step = fl(step - fl(1-step));
//   endpoint is exactly 1.0 (JAX concatenates the stop value).
__device__ __forceinline__ float grid_coord(int v) {
    if (v == H_DIM - 1) return 1.0f;
    float s = __fdiv_rn((float)v, SCALE);
    float a = __fsub_rn(1.0f, s);
    return __fsub_rn(s, a);
}

// Bit-exact mapping chain: x = (0.5*((g - d) + 1.0)) * 159, each op fp32 RN,
// no fma contraction (round() anchor is discontinuous -> ulps matter).
__device__ __forceinline__ float map_coord(float g, float d) {
    float t = __fsub_rn(g, d);
    float u = __fadd_rn(t, 1.0f);
    float v = __fmul_rn(0.5f, u);
    return __fmul_rn(v, SCALE);
}

__device__ __forceinline__ int clampi(int v) {
    v = v < 0 ? 0 : v;
    return v > MAXC ? MAXC : v;
}

__global__ void __launch_bounds__(H_DIM)
warp3d_kernel(const float* __restrict__ img,
              const float* __restrict__ disp,
              float* __restrict__ out) {
    const int k = threadIdx.x;        // D (fastest)
    const int j = blockIdx.x;         // W
    const int i = blockIdx.y;         // H
    const int b = blockIdx.z;         // B

    const int base_b = b * (H_DIM * H_DIM * H_DIM);
    const int idx = base_b + (i * H_DIM + j) * H_DIM + k;

    // Prefetch the identity-position img cacheline (gfx1250 global_prefetch_b8).
    // The 8 gathers land within ~±4 voxels in z / nearby lines in x,y; the
    // coordinate dependency chain below gives the prefetch time to land.
    __builtin_prefetch(&img[idx], 0, 3);

    // disp: contiguous float3 per voxel -> b96-class per-lane load.
    const float* dp = disp + (size_t)idx * 3;
    float dx = dp[0];
    float dy = dp[1];
    float dz = dp[2];

    // grid_coord(i), grid_coord(j) are wave-uniform (block-constant).
    float x = map_coord(grid_coord(i), dx);
    float y = map_coord(grid_coord(j), dy);
    float z = map_coord(grid_coord(k), dz);

    // round-half-even anchor == jnp.round (v_rndne_f32); x1 clips (x0+1)
    // of the *unclipped* anchor, matching the reference order.
    int x0 = (int)rintf(x);
    int y0 = (int)rintf(y);
    int z0 = (int)rintf(z);
    int x1 = clampi(x0 + 1);
    int y1 = clampi(y0 + 1);
    int z1 = clampi(z0 + 1);
    x0 = clampi(x0);
    y0 = clampi(y0);
    z0 = clampi(z0);

    // Shared address terms for the 8 gathers.
    const int rx0 = x0 * (H_DIM * H_DIM), rx1 = x1 * (H_DIM * H_DIM);
    const int cy0 = y0 * H_DIM,           cy1 = y1 * H_DIM;
    const bool inx0 = x0 < H_DIM, inx1 = x1 < H_DIM;
    const bool iny0 = y0 < H_DIM, iny1 = y1 < H_DIM;
    const bool inz0 = z0 < H_DIM, inz1 = z1 < H_DIM;

    const float* p = img + base_b;
    float I000 = (inx0 && iny0 && inz0) ? p[rx0 + cy0 + z0] : 0.0f;
    float I001 = (inx0 && iny0 && inz1) ? p[rx0 + cy0 + z1] : 0.0f;
    float I010 = (inx0 && iny1 && inz0) ? p[rx0 + cy1 + z0] : 0.0f;
    float I100 = (inx1 && iny0 && inz0) ? p[rx1 + cy0 + z0] : 0.0f;
    float I011 = (inx0 && iny1 && inz1) ? p[rx0 + cy1 + z1] : 0.0f;
    float I101 = (inx1 && iny0 && inz1) ? p[rx1 + cy0 + z1] : 0.0f;
    float I110 = (inx1 && iny1 && inz0) ? p[rx1 + cy1 + z0] : 0.0f;
    float I111 = (inx1 && iny1 && inz1) ? p[rx1 + cy1 + z1] : 0.0f;

    float x0f = (float)x0, x1f = (float)x1;
    float y0f = (float)y0, y1f = (float)y1;
    float z0f = (float)z0, z1f = (float)z1;

    float wx0 = x1f - x, wx1 = x - x0f;   // may exceed [0,1] by design (round anchor)
    float wy0 = y1f - y, wy1 = y - y0f;
    float wz0 = z1f - z, wz1 = z - z0f;

    float W000 = (wx0 * wy0) * wz0;
    float W001 = (wx0 * wy0) * wz1;
    float W010 = (wx0 * wy1) * wz0;
    float W100 = (wx1 * wy0) * wz0;
    float W011 = (wx0 * wy1) * wz1;
    float W101 = (wx1 * wy0) * wz1;
    float W110 = (wx1 * wy1) * wz0;
    float W111 = (wx1 * wy1) * wz1;

    float acc = W000 * I000 + W001 * I001 +
                W010 * I010 + W100 * I100 +
                W011 * I011 + W101 * I101 +
                W110 * I110 + W111 * I111;

    out[idx] = acc;
}

extern "C" void kernel_launch(void* const* d_in, const int* in_sizes, int n_in,
                              void* d_out, int out_size, void* d_ws, size_t ws_size,
                              hipStream_t stream) {
    const float* img  = (const float*)d_in[0];  // (2,160,160,160,1)
    const float* disp = (const float*)d_in[1];  // (2,160,160,160,3)
    float* out = (float*)d_out;                 // (2,160,160,160,1)

    dim3 grid(H_DIM, H_DIM, 2);   // (W, H, B)
    dim3 block(H_DIM);            // one z-row = 5 wave32
    warp3d_kernel<<<grid, block, 0, stream>>>(img, disp, out);
}